// SuperFusion_29832842838034
// MI455X (gfx1250) — compile-verified
//
#include <hip/hip_runtime.h>
#include <hip/hip_bf16.h>

// LSS voxel pooling: out[b, c, ix, iy] += x[p, c] for (ix, iy, b) = geom[p].
// Shapes (from reference setup): x[692736, 64] f32, geom[692736, 4] i32,
// out[4, 64, 256, 256] f32 (nz = 1 collapses away; gz = 0).
//
// out flat index = ((b*64 + c)*256 + ix)*256 + iy
//                = b*4194304 + c*65536 + ix*256 + iy
//
// Roofline: ~177 MB x + 11 MB geom streamed once (NT-hinted), 64 MB zero-fill
// + 44M f32 atomics that resolve in the 192 MB L2 (output stays resident).
// ~14 us floor at 23.3 TB/s. No matmul -> WMMA does not apply; the CDNA5
// paths used are the async global->LDS engine (ASYNCcnt), wave32 scalar
// address formation, and native global_atomic_add_f32.

#define CFEAT          64
#define NXG            256
#define NYG            256
#define PLANE          (NXG * NYG)      // 65536
#define OUTPB          (CFEAT * PLANE)  // 4194304
#define PTS_PER_BLOCK  8                // 8 wave32 waves = 256 threads

// gfx1250 async global->LDS copy engine (ASYNCcnt path, CDNA5 ISA).
#if defined(__AMDGCN__) && __has_builtin(__builtin_amdgcn_global_load_async_to_lds_b64)
#define LSS_USE_ASYNC_LDS 1
#else
#define LSS_USE_ASYNC_LDS 0
#endif

typedef int lss_v4i __attribute__((ext_vector_type(4)));
#if LSS_USE_ASYNC_LDS
typedef int lss_v2i __attribute__((ext_vector_type(2)));
typedef lss_v2i __attribute__((address_space(1)))* lss_gv2i_ptr;  // global
typedef lss_v2i __attribute__((address_space(3)))* lss_lv2i_ptr;  // LDS
// CPol (gfx12+/gfx1250): TH in bits [2:0]; TH=1 -> NT (non-temporal).
#define LSS_CPOL_NT 1
#endif

__global__ __launch_bounds__(256) void lss_zero_f4(float4* __restrict__ out, int n4) {
  int i = blockIdx.x * 256 + threadIdx.x;
  if (i < n4) out[i] = make_float4(0.f, 0.f, 0.f, 0.f);
}

__global__ __launch_bounds__(256) void lss_voxel_scatter(
    const float* __restrict__ x,
    const int*   __restrict__ geom,
    float*       __restrict__ out,
    int n_points) {
#if LSS_USE_ASYNC_LDS
  __shared__ float smem[PTS_PER_BLOCK * CFEAT];
#endif
  const int lane = threadIdx.x & 31;   // wave32 lane
  const int wib  = threadIdx.x >> 5;   // wave-in-block: 0..7
  const int p    = blockIdx.x * PTS_PER_BLOCK + wib;  // point id, uniform per wave
  if (p >= n_points) return;           // whole wave exits together (p is wave-uniform)

#if LSS_USE_ASYNC_LDS
  {
    // Each lane async-copies 8 bytes: the wave moves the point's full 256 B row
    // into its private LDS segment with one instruction, tracked by ASYNCcnt.
    // NT hint: x is streamed exactly once; keep L2 free for the output atomics.
    const float* gsrc = x + (size_t)p * CFEAT + (size_t)(lane * 2);
    float*       ldst = &smem[wib * CFEAT + lane * 2];
    __builtin_amdgcn_global_load_async_to_lds_b64(
        (lss_gv2i_ptr)(void*)gsrc,
        (lss_lv2i_ptr)(void*)ldst,
        /*imm offset=*/0, /*cpol=*/LSS_CPOL_NT);
  }
#endif

  // Per-point voxel coords are wave-uniform: force them into SGPRs so the
  // output base address is scalar (wave32 scalarization). NT load: geom is
  // streamed once.
  const lss_v4i g = __builtin_nontemporal_load((const lss_v4i*)geom + p);
  const int  gx = __builtin_amdgcn_readfirstlane(g.x);
  const int  gy = __builtin_amdgcn_readfirstlane(g.y);
  const int  gb = __builtin_amdgcn_readfirstlane(g.w);
  const size_t base = (size_t)gb * OUTPB + (size_t)gx * NYG + (size_t)gy;

#if LSS_USE_ASYNC_LDS
#if __has_builtin(__builtin_amdgcn_s_wait_asynccnt)
  __builtin_amdgcn_s_wait_asynccnt(0);
#else
  asm volatile("s_wait_asynccnt 0" ::: "memory");
#endif
  const float v0 = smem[wib * CFEAT + lane];        // conflict-free ds_load
  const float v1 = smem[wib * CFEAT + lane + 32];
#else
  const float v0 = x[(size_t)p * CFEAT + lane];
  const float v1 = x[(size_t)p * CFEAT + lane + 32];
#endif

  // Lane l accumulates channels l and l+32: no-return global_atomic_add_f32,
  // resolved in L2 (64 MB output << 192 MB L2).
  atomicAdd(out + base + (size_t)lane * PLANE, v0);
  atomicAdd(out + base + (size_t)(lane + 32) * PLANE, v1);
}

extern "C" void kernel_launch(void* const* d_in, const int* in_sizes, int n_in,
                              void* d_out, int out_size, void* d_ws, size_t ws_size,
                              hipStream_t stream) {
  const float* x    = (const float*)d_in[0];
  const int*   geom = (const int*)d_in[1];
  float*       out  = (float*)d_out;

  const int n_points = in_sizes[0] / CFEAT;   // 692736
  const int n4       = out_size / 4;          // out_size = 16777216 floats

  // Must zero every call: atomics accumulate and the harness replays the graph.
  lss_zero_f4<<<dim3((n4 + 255) / 256), dim3(256), 0, stream>>>((float4*)out, n4);

  lss_voxel_scatter<<<dim3((n_points + PTS_PER_BLOCK - 1) / PTS_PER_BLOCK),
                      dim3(256), 0, stream>>>(x, geom, out, n_points);
}